// GIN_214748365115
// MI455X (gfx1250) — compile-verified
//
#include <hip/hip_runtime.h>

typedef float v2f __attribute__((ext_vector_type(2)));
typedef float v8f __attribute__((ext_vector_type(8)));

#define D64 64
#define NCLS 10
#define BN_EPS 1e-5f

// ---------------------------------------------------------------------------
// zero a float region
__global__ void gin_zero(float* __restrict__ p, long long n) {
    long long i = (long long)blockIdx.x * blockDim.x + threadIdx.x;
    if (i < n) p[i] = 0.0f;
}

// agg = (1 + eps[l]) * h   (elementwise)
__global__ void gin_scale_copy(float* __restrict__ out, const float* __restrict__ in,
                               const float* __restrict__ eps, int l, long long n) {
    long long i = (long long)blockIdx.x * blockDim.x + threadIdx.x;
    if (i >= n) return;
    float s = 1.0f + eps[l];
    out[i] = s * in[i];
}

// agg[dst] += h[src]  — one thread per (edge, 4-float chunk)
__global__ void gin_scatter(float* __restrict__ agg, const float* __restrict__ h,
                            const int* __restrict__ src, const int* __restrict__ dst,
                            int E) {
    long long tid = (long long)blockIdx.x * blockDim.x + threadIdx.x;
    long long e = tid >> 4;
    if (e >= E) return;
    int f = (int)(tid & 15) * 4;
    int s = src[e], d = dst[e];
    const float4 v = *(const float4*)(h + (size_t)s * D64 + f);
    float* p = agg + (size_t)d * D64 + f;
    atomicAdd(p + 0, v.x);
    atomicAdd(p + 1, v.y);
    atomicAdd(p + 2, v.z);
    atomicAdd(p + 3, v.w);
}

// out[N,64] = in[N,64] @ W[64,64] + bias[64]   — f32 WMMA 16x16x4
// 256 threads = 8 waves; wave w: rows block*32 + (w>>2)*16, cols (w&3)*16
__global__ __launch_bounds__(256) void gin_gemm64(const float* __restrict__ in,
                                                  const float* __restrict__ W,
                                                  const float* __restrict__ bias,
                                                  float* __restrict__ out, int nrows) {
    const int lane = threadIdx.x & 31;
    const int wave = threadIdx.x >> 5;
    const int half = lane >> 4;       // 0/1: selects K pair
    const int lm   = lane & 15;
    const int rowTile = blockIdx.x * 32 + (wave >> 2) * 16;
    const int c0      = (wave & 3) * 16;
    const int colB    = c0 + lm;

    // C/D layout: VGPR j, lanes 0-15 -> row rowTile+j, lanes 16-31 -> row rowTile+j+8
    v8f acc;
    float bv = bias[colB];
#pragma unroll
    for (int j = 0; j < 8; ++j) acc[j] = bv;

    int mrow = rowTile + lm;
    if (mrow >= nrows) mrow = nrows - 1;              // clamp (keeps EXEC full)
    const float* arow = in + (size_t)mrow * D64 + half * 2;

#pragma unroll
    for (int kk = 0; kk < 16; ++kk) {
        const int kb = kk * 4;
        v2f a, b;
        // A 16x4 frag: lane holds A[m][kb + 2*half + v]
        a.x = arow[kb + 0];
        a.y = arow[kb + 1];
        // B 4x16 frag: lane holds W[kb + 2*half + v][colB]
        b.x = W[(size_t)(kb + half * 2 + 0) * D64 + colB];
        b.y = W[(size_t)(kb + half * 2 + 1) * D64 + colB];
        acc = __builtin_amdgcn_wmma_f32_16x16x4_f32(false, a, false, b,
                                                    (short)0, acc, false, false);
    }

#pragma unroll
    for (int j = 0; j < 8; ++j) {
        int row = rowTile + j + half * 8;
        if (row < nrows) out[(size_t)row * D64 + colB] = acc[j];
    }
}

// per-feature sum and sum-of-squares -> stats[0:64]=sum, stats[64:128]=sumsq
__global__ void gin_stats(const float* __restrict__ in, int nrows,
                          float* __restrict__ stats) {
    int tid = blockIdx.x * blockDim.x + threadIdx.x;
    int f = tid & 63;
    int r = tid >> 6;
    int stride = (gridDim.x * blockDim.x) >> 6;
    float s = 0.0f, ss = 0.0f;
    for (int row = r; row < nrows; row += stride) {
        float v = in[(size_t)row * D64 + f];
        s += v;
        ss += v * v;
    }
    atomicAdd(&stats[f], s);
    atomicAdd(&stats[D64 + f], ss);
}

// out = relu(gamma * (in - mean) * rsqrt(var + eps) + beta)
__global__ void gin_bn_relu(float* __restrict__ out, const float* __restrict__ in,
                            const float* __restrict__ stats,
                            const float* __restrict__ gamma,
                            const float* __restrict__ beta, int nrows) {
    long long i = (long long)blockIdx.x * blockDim.x + threadIdx.x;
    long long n = (long long)nrows * D64;
    if (i >= n) return;
    int f = (int)(i & 63);
    float invN = 1.0f / (float)nrows;
    float m = stats[f] * invN;
    float var = stats[D64 + f] * invN - m * m;   // biased variance
    float rs = rsqrtf(var + BN_EPS);
    float v = gamma[f] * (in[i] - m) * rs + beta[f];
    out[i] = v > 0.0f ? v : 0.0f;
}

// pooled[batch[i]] += h[i]  — one thread per (node, 4-float chunk)
__global__ void gin_pool(float* __restrict__ pooled, const float* __restrict__ h,
                         const int* __restrict__ batch, int nrows) {
    long long tid = (long long)blockIdx.x * blockDim.x + threadIdx.x;
    long long node = tid >> 4;
    if (node >= nrows) return;
    int f = (int)(tid & 15) * 4;
    int g = batch[node];
    const float4 v = *(const float4*)(h + (size_t)node * D64 + f);
    float* p = pooled + (size_t)g * D64 + f;
    atomicAdd(p + 0, v.x);
    atomicAdd(p + 1, v.y);
    atomicAdd(p + 2, v.z);
    atomicAdd(p + 3, v.w);
}

// score[g,c] = sum_l ( pooled[l,g,:] @ Wp[l,:,c] + bp[l,c] )
__global__ void gin_score(float* __restrict__ score, const float* __restrict__ pooled,
                          const float* __restrict__ Wp, const float* __restrict__ bp,
                          int G, int nl) {
    int tid = blockIdx.x * blockDim.x + threadIdx.x;
    if (tid >= G * NCLS) return;
    int g = tid / NCLS, c = tid % NCLS;
    float acc = 0.0f;
    for (int l = 0; l < nl; ++l) {
        acc += bp[l * NCLS + c];
        const float* pl = pooled + ((size_t)l * G + g) * D64;
        const float* wl = Wp + (size_t)l * D64 * NCLS + c;
        for (int k = 0; k < D64; ++k) acc += pl[k] * wl[k * NCLS];
    }
    score[tid] = acc;
}

// ---------------------------------------------------------------------------
extern "C" void kernel_launch(void* const* d_in, const int* in_sizes, int n_in,
                              void* d_out, int out_size, void* d_ws, size_t ws_size,
                              hipStream_t stream) {
    const float* x     = (const float*)d_in[0];
    const int*   ei    = (const int*)d_in[1];
    const int*   batch = (const int*)d_in[2];
    const float* eps   = (const float*)d_in[3];
    const float* W1    = (const float*)d_in[4];
    const float* b1    = (const float*)d_in[5];
    const float* g1    = (const float*)d_in[6];
    const float* be1   = (const float*)d_in[7];
    const float* W2    = (const float*)d_in[8];
    const float* b2    = (const float*)d_in[9];
    const float* gout  = (const float*)d_in[10];
    const float* beout = (const float*)d_in[11];
    const float* Wp    = (const float*)d_in[12];
    const float* bp    = (const float*)d_in[13];

    const int N  = in_sizes[0] / D64;
    const int E  = in_sizes[1] / 2;
    const int Lc = in_sizes[3];            // GIN layers (4)
    const int G  = out_size / NCLS;        // graphs (512)

    const int* src = ei;
    const int* dst = ei + E;

    const long long ND = (long long)N * D64;
    float* ws      = (float*)d_ws;
    float* hbuf    = ws;
    float* aggbuf  = hbuf + ND;
    float* tbuf    = aggbuf + ND;
    float* pooled  = tbuf + ND;                        // (Lc+1)*G*64
    float* stats   = pooled + (size_t)(Lc + 1) * G * D64;  // Lc*2*128

    const int TB = 256;
    const long long zeroN = (long long)(Lc + 1) * G * D64 + (long long)Lc * 2 * 128;
    gin_zero<<<(unsigned)((zeroN + TB - 1) / TB), TB, 0, stream>>>(ws + 3 * ND, zeroN);

    // pool the raw input features (hidden_rep[0] = x)
    {
        long long t = (long long)N * 16;
        gin_pool<<<(unsigned)((t + TB - 1) / TB), TB, 0, stream>>>(pooled, x, batch, N);
    }

    const unsigned gElem    = (unsigned)((ND + TB - 1) / TB);
    const unsigned gScatter = (unsigned)(((long long)E * 16 + TB - 1) / TB);
    const unsigned gGemm    = (unsigned)((N + 31) / 32);
    const unsigned gPool    = (unsigned)(((long long)N * 16 + TB - 1) / TB);

    const float* hin = x;
    for (int l = 0; l < Lc; ++l) {
        float* st1 = stats + (size_t)l * 256;
        float* st2 = st1 + 128;

        // GINConv aggregation: agg = (1+eps)*h + scatter_add(h[src] -> dst)
        gin_scale_copy<<<gElem, TB, 0, stream>>>(aggbuf, hin, eps, l, ND);
        gin_scatter<<<gScatter, TB, 0, stream>>>(aggbuf, hin, src, dst, E);

        // MLP linear 1 + BN + ReLU
        gin_gemm64<<<gGemm, TB, 0, stream>>>(aggbuf, W1 + (size_t)l * D64 * D64,
                                             b1 + (size_t)l * D64, tbuf, N);
        gin_stats<<<1024, TB, 0, stream>>>(tbuf, N, st1);
        gin_bn_relu<<<gElem, TB, 0, stream>>>(aggbuf, tbuf, st1,
                                              g1 + (size_t)l * D64,
                                              be1 + (size_t)l * D64, N);

        // MLP linear 2 + outer BN + ReLU
        gin_gemm64<<<gGemm, TB, 0, stream>>>(aggbuf, W2 + (size_t)l * D64 * D64,
                                             b2 + (size_t)l * D64, tbuf, N);
        gin_stats<<<1024, TB, 0, stream>>>(tbuf, N, st2);
        gin_bn_relu<<<gElem, TB, 0, stream>>>(hbuf, tbuf, st2,
                                              gout + (size_t)l * D64,
                                              beout + (size_t)l * D64, N);

        // jumping-knowledge pooling of this layer's representation
        gin_pool<<<gPool, TB, 0, stream>>>(pooled + (size_t)(l + 1) * G * D64,
                                           hbuf, batch, N);
        hin = hbuf;
    }

    // readout: score[g,c]
    {
        int t = G * NCLS;
        gin_score<<<(t + TB - 1) / TB, TB, 0, stream>>>((float*)d_out, pooled,
                                                        Wp, bp, G, Lc + 1);
    }
}